// _ConvCaps_79267916415234
// MI455X (gfx1250) — compile-verified
//
#include <hip/hip_runtime.h>

// CDNA5 / gfx1250: wave32, WMMA f32 16x16x4 (full-f32 matrix core path).
typedef __attribute__((ext_vector_type(2))) float v2f;
typedef __attribute__((ext_vector_type(8))) float v8f;

#define CAPS_L   32       // in capsules
#define CAPS_O   32       // out capsules
#define HW       196      // 14*14 spatial positions
#define CIN      544      // L*16 + L channels
#define TILE_S   20       // LDS row stride (floats): 16B aligned, bank-spread
#define NWAVE    8        // 256 threads / wave32

__global__ __launch_bounds__(256)
void _ConvCaps_spectral_kernel(const float* __restrict__ in,
                               const float* __restrict__ wts,
                               const float* __restrict__ bias,
                               float* __restrict__ out)
{
    // per-wave vote tiles: 32 rows (n) x 16 cols (p), stride 20 floats
    __shared__ float tile[NWAVE * 32 * TILE_S];   // 20480 B
    __shared__ float red[NWAVE * 256];            // 8192 B  cross-wave Gram reduce
    __shared__ float Gs[16 * 16];                 // Gram matrix
    __shared__ float ubuf[16];
    __shared__ float ybuf[16];

    const int tid  = threadIdx.x;
    const int lane = tid & 31;
    const int wave = tid >> 5;
    const int bo   = blockIdx.x;          // b*O + o
    const int b    = bo >> 5;
    const int o    = bo & 31;

    // Per-lane weight block: lane l owns W[l, o, :, :] (16 f32, stays in VGPRs)
    float wr[16];
    {
        const float4* wp4 = (const float4*)(wts + (size_t)(lane * CAPS_O + o) * 16);
        #pragma unroll
        for (int i = 0; i < 4; ++i) {
            float4 t = wp4[i];
            wr[4*i+0] = t.x; wr[4*i+1] = t.y; wr[4*i+2] = t.z; wr[4*i+3] = t.w;
        }
    }

    float* mytile = &tile[wave * (32 * TILE_S)];
    const float* binp = in + (size_t)b * HW * CIN;

    const int col  = lane & 15;           // WMMA fragment column (M for A / N for B)
    const int krow = (lane >> 4) << 1;    // K base: 0 for lanes 0-15, 2 for 16-31

    v8f acc = {};                         // 16x16 f32 Gram accumulator (8 VGPRs)

    for (int hw = wave; hw < HW; hw += NWAVE) {
        const float* pp = binp + (size_t)hw * CIN + lane * 16;
        if (hw + NWAVE < HW)
            __builtin_prefetch(binp + (size_t)(hw + NWAVE) * CIN + lane * 16, 0, 1);

        // lane l: pose of child capsule l at this position (4x4, row-major)
        float4 p0 = ((const float4*)pp)[0];
        float4 p1 = ((const float4*)pp)[1];
        float4 p2 = ((const float4*)pp)[2];
        float4 p3 = ((const float4*)pp)[3];
        float  a  = binp[(size_t)hw * CIN + CAPS_L * 16 + lane];  // child activation

        float pr[4][4] = {{p0.x,p0.y,p0.z,p0.w},
                          {p1.x,p1.y,p1.z,p1.w},
                          {p2.x,p2.y,p2.z,p2.w},
                          {p3.x,p3.y,p3.z,p3.w}};

        // vote v = a * (pose @ W), flattened (i,k) -> i*4+k
        float v[16];
        #pragma unroll
        for (int i = 0; i < 4; ++i) {
            #pragma unroll
            for (int k = 0; k < 4; ++k) {
                v[i*4+k] = a * (pr[i][0]*wr[0*4+k] + pr[i][1]*wr[1*4+k] +
                                pr[i][2]*wr[2*4+k] + pr[i][3]*wr[3*4+k]);
            }
        }

        // stage V[n=lane][p=0..15] into this wave's LDS tile (16B-aligned b128 stores)
        float4* dst = (float4*)&mytile[lane * TILE_S];
        dst[0] = make_float4(v[0],  v[1],  v[2],  v[3]);
        dst[1] = make_float4(v[4],  v[5],  v[6],  v[7]);
        dst[2] = make_float4(v[8],  v[9],  v[10], v[11]);
        dst[3] = make_float4(v[12], v[13], v[14], v[15]);

        // G += V^T V over the 32 rows, K=4 per WMMA step.
        // A[p,k]=V[n0+k][p], B[k,q]=V[n0+k][q]: for V^T V the A and B lane
        // fragments are identical -> pass the same register pair twice.
        #pragma unroll
        for (int s = 0; s < 8; ++s) {
            int r = s * 4 + krow;
            v2f ab;
            ab.x = mytile[r * TILE_S + col];        // K = krow
            ab.y = mytile[(r + 1) * TILE_S + col];  // K = krow + 1
            acc = __builtin_amdgcn_wmma_f32_16x16x4_f32(
                      false, ab, false, ab, (short)0, acc, false, false);
        }
    }

    // ---- cross-wave reduction of the 8 partial Gram accumulators ----
    #pragma unroll
    for (int r = 0; r < 8; ++r) red[wave * 256 + r * 32 + lane] = acc[r];
    __syncthreads();
    {
        int r  = tid >> 5;                // accumulator VGPR index 0..7
        int ln = tid & 31;
        float g = 0.f;
        #pragma unroll
        for (int w8 = 0; w8 < NWAVE; ++w8) g += red[w8 * 256 + r * 32 + ln];
        // C/D layout: M = r + 8*(lane/16), N = lane%16
        int m = r + ((ln >> 4) << 3);
        int n = ln & 15;
        Gs[m * 16 + n] = g;
    }
    __syncthreads();

    // ---- power iteration on 16x16 PSD Gram (replaces eigh top-vector) ----
    if (tid < 16) ubuf[tid] = 0.25f;
    __syncthreads();
    for (int it = 0; it < 40; ++it) {
        if (tid < 16) {
            float y = 0.f;
            #pragma unroll
            for (int n = 0; n < 16; ++n) y += Gs[tid * 16 + n] * ubuf[n];
            ybuf[tid] = y;
        }
        __syncthreads();
        if (tid < 16) {
            float s = 0.f;
            #pragma unroll
            for (int n = 0; n < 16; ++n) { float t = ybuf[n]; s += t * t; }
            ubuf[tid] = ybuf[tid] * rsqrtf(fmaxf(s, 1e-30f));
        }
        __syncthreads();
    }

    // lambda_max = u^T G u = yu_norm_sq ; trace(G) = y_fro_sq
    if (tid == 0) {
        float lam = 0.f, tr = 0.f;
        for (int m = 0; m < 16; ++m) {
            float y = 0.f;
            for (int n = 0; n < 16; ++n) y += Gs[m * 16 + n] * ubuf[n];
            lam += y * ubuf[m];
            tr  += Gs[m * 16 + m];
        }
        float x = lam / tr - bias[o];
        out[bo] = 1.f / (1.f + __expf(-x));
    }
    if (tid < 16) out[512 + bo * 16 + tid] = ubuf[tid];   // poses [B,O,16]
}

extern "C" void kernel_launch(void* const* d_in, const int* in_sizes, int n_in,
                              void* d_out, int out_size, void* d_ws, size_t ws_size,
                              hipStream_t stream) {
    (void)in_sizes; (void)n_in; (void)out_size; (void)d_ws; (void)ws_size;
    const float* in   = (const float*)d_in[0];   // [16,14,14,544]
    const float* wts  = (const float*)d_in[1];   // [1,32,32,4,4]
    const float* bias = (const float*)d_in[2];   // [32]
    float* out = (float*)d_out;                  // 512 activations + 8192 poses
    _ConvCaps_spectral_kernel<<<dim3(16 * 32), dim3(256), 0, stream>>>(in, wts, bias, out);
}